// filterbank_31782757991044
// MI455X (gfx1250) — compile-verified
//
#include <hip/hip_runtime.h>
#include <stdint.h>

typedef __attribute__((ext_vector_type(2))) float v2f;
typedef __attribute__((ext_vector_type(4))) float v4f;
typedef __attribute__((ext_vector_type(8))) float v8f;

#define NF   9
#define TT   1000
#define LBLK 16
#define NBLK 63   // ceil(1000/16), last block has 8 valid samples

__global__ __launch_bounds__(288) void fb_wmma_kernel(
    const float* __restrict__ x,   // [4096][1000]
    const float* __restrict__ Bc,  // [9][K]
    const float* __restrict__ Ac,  // [9][K]
    float* __restrict__ out,       // [4096][1000][9]
    int K)
{
    __shared__ __align__(16) float sh[14400]; // 57.6 KB, phase1/phase2 overlaid
    const int tid = threadIdx.x;
    const int w   = tid >> 5;   // wave id == filter id (0..8)
    const int l   = tid & 31;
    const int m16 = l & 15;
    const int half = l >> 4;
    const int kb  = half * 2;
    const int seq_base = blockIdx.x * 16;

    // ---------------- Phase 1: build per-filter block matrices in LDS ----------
    float* Hm = sh + w * 1600;   // [16][16] Toeplitz impulse response
    float* Wm = Hm + 256;        // [16][16] rows = e0^T A^j
    float* Gm = Wm + 256;        // [16][16] cols = A^(15-i) g
    float* Pm = Gm + 256;        // [16][16] A^16
    float* Am = Pm + 256;        // scratch: A
    float* Tm = Am + 256;        // scratch
    float* gv = Tm + 256;        // [16]
    float* hv = gv + 16;         // [16]
    float* bn = hv + 16;         // [16]
    float* an = bn + 16;         // [16]

    int n = K - 1; if (n > 15) n = 15;

    if (l < 16) {
        bn[l] = (l < K) ? Bc[w * K + l] : 0.f;
        an[l] = (l < K) ? Ac[w * K + l] : 0.f;
    }
    __syncthreads();
    {
        float inva = 1.f / an[0];
        __syncthreads();
        if (l < 16) { bn[l] *= inva; an[l] *= inva; }
    }
    __syncthreads();

    // A (companion-ish, zero-padded) and g = b[1:] - a[1:]*b0
    #pragma unroll
    for (int it = 0; it < 8; ++it) {
        int idx = l + 32 * it;
        int r = idx >> 4, c0 = idx & 15;
        float v = 0.f;
        if (r < n - 1 && c0 == r + 1) v += 1.f;
        if (r < n && c0 == 0)         v -= an[r + 1];
        Am[idx] = v;
    }
    if (l < 16) gv[l] = (l < n) ? (bn[l + 1] - an[l + 1] * bn[0]) : 0.f;
    __syncthreads();

    // W rows: W[0]=e0; W[j] = W[j-1] * A
    if (l < 16) Wm[l] = (l == 0) ? 1.f : 0.f;
    __syncthreads();
    for (int j = 1; j < 16; ++j) {
        float s = 0.f;
        if (l < 16)
            for (int i = 0; i < 16; ++i) s += Wm[(j - 1) * 16 + i] * Am[i * 16 + l];
        __syncthreads();
        if (l < 16) Wm[j * 16 + l] = s;
        __syncthreads();
    }
    // impulse response h[0]=b0, h[m]=W[m-1].g ; H = lower-tri Toeplitz
    if (l == 0) hv[0] = bn[0];
    if (l >= 1 && l < 16) {
        float s = 0.f;
        for (int i = 0; i < 16; ++i) s += Wm[(l - 1) * 16 + i] * gv[i];
        hv[l] = s;
    }
    __syncthreads();
    #pragma unroll
    for (int it = 0; it < 8; ++it) {
        int idx = l + 32 * it; int r = idx >> 4, c0 = idx & 15;
        Hm[idx] = (r >= c0) ? hv[r - c0] : 0.f;
    }
    // Pm = A^16 via 4 squarings
    #pragma unroll
    for (int it = 0; it < 8; ++it) { int idx = l + 32 * it; Pm[idx] = Am[idx]; }
    __syncthreads();
    for (int sq = 0; sq < 4; ++sq) {
        #pragma unroll
        for (int it = 0; it < 8; ++it) {
            int idx = l + 32 * it; int r = idx >> 4, c0 = idx & 15;
            float s = 0.f;
            for (int i = 0; i < 16; ++i) s += Pm[r * 16 + i] * Pm[i * 16 + c0];
            Tm[idx] = s;
        }
        __syncthreads();
        #pragma unroll
        for (int it = 0; it < 8; ++it) { int idx = l + 32 * it; Pm[idx] = Tm[idx]; }
        __syncthreads();
    }
    // G columns: v=g -> col15; v = A v -> col 14.. (ping-pong in Tm[0..31])
    if (l < 16) { Tm[l] = gv[l]; Gm[l * 16 + 15] = gv[l]; }
    __syncthreads();
    for (int i2 = 14; i2 >= 0; --i2) {
        int pp = (14 - i2) & 1;
        float s = 0.f;
        if (l < 16)
            for (int jj = 0; jj < 16; ++jj) s += Am[l * 16 + jj] * Tm[pp * 16 + jj];
        __syncthreads();
        if (l < 16) { Tm[(1 - pp) * 16 + l] = s; Gm[l * 16 + i2] = s; }
        __syncthreads();
    }

    // Load constants into A-matrix fragments (16x4 f32: lane=M, K-pair=(lane>>4)*2)
    v2f Hreg[4], Wreg[4], Greg[4], Preg[4];
    #pragma unroll
    for (int c = 0; c < 4; ++c) {
        Hreg[c] = *(const v2f*)&Hm[m16 * 16 + 4 * c + kb];
        Wreg[c] = *(const v2f*)&Wm[m16 * 16 + 4 * c + kb];
        Greg[c] = *(const v2f*)&Gm[m16 * 16 + 4 * c + kb];
        Preg[c] = *(const v2f*)&Pm[m16 * 16 + 4 * c + kb];
    }
    __syncthreads();  // phase-1 LDS dead from here

    // ---------------- Phase 2: blocked scan with WMMA -------------------------
    float* zt  = sh;               // [9][16 seq][16 state]
    float* yt  = sh + 9 * 256;     // [16 seq][16 t][9 f]
    float* xtA = yt + 2304;        // [16 seq][16 t]
    float* xtB = xtA + 256;
    float* ztw = zt + w * 256;

    {   // zero own state tile (per-wave private; in-order within wave)
        v4f z0 = {0.f, 0.f, 0.f, 0.f};
        ((v4f*)ztw)[l * 2]     = z0;
        ((v4f*)ztw)[l * 2 + 1] = z0;
    }

    auto issue_tile = [&](int blk, float* xt) {  // wave 0 only
        bool partial = (blk == NBLK - 1);
        const float* gp = x + (size_t)(seq_base + m16) * TT + blk * LBLK + half * 8;
        uint32_t lo = (uint32_t)(uintptr_t)&xt[m16 * 16 + half * 8];
        if (!partial || half == 0) {
            asm volatile("global_load_async_to_lds_b128 %0, %1, off\n\t"
                         "global_load_async_to_lds_b128 %0, %1, off offset:16"
                         :: "v"(lo), "v"(gp) : "memory");
        }
        if (partial && half == 0) {  // zero t=8..15 tail (disjoint from async dest)
            v4f z0 = {0.f, 0.f, 0.f, 0.f};
            *(v4f*)&xt[m16 * 16 + 8]  = z0;
            *(v4f*)&xt[m16 * 16 + 12] = z0;
        }
    };

    if (w == 0) {
        issue_tile(0, xtA);
        asm volatile("s_wait_asynccnt 0x0" ::: "memory");
    }

    for (int blk = 0; blk < NBLK; ++blk) {
        __syncthreads();  // tile[blk] ready; previous ytile copy complete
        float* xt = (blk & 1) ? xtB : xtA;
        if (w == 0 && blk + 1 < NBLK) issue_tile(blk + 1, (blk & 1) ? xtA : xtB);

        // B-matrix fragments (4x16 f32: lane=N, K-pair=(lane>>4)*2)
        v2f Xr[4], Zr[4];
        #pragma unroll
        for (int c = 0; c < 4; ++c) {
            Xr[c] = *(const v2f*)&xt [m16 * 16 + 4 * c + kb];
            Zr[c] = *(const v2f*)&ztw[m16 * 16 + 4 * c + kb];
        }

        v8f y  = {0.f,0.f,0.f,0.f,0.f,0.f,0.f,0.f};
        v8f zn = {0.f,0.f,0.f,0.f,0.f,0.f,0.f,0.f};
        #pragma unroll
        for (int c = 0; c < 4; ++c)
            y  = __builtin_amdgcn_wmma_f32_16x16x4_f32(false, Hreg[c], false, Xr[c], (short)0, y,  false, false);
        #pragma unroll
        for (int c = 0; c < 4; ++c)
            y  = __builtin_amdgcn_wmma_f32_16x16x4_f32(false, Wreg[c], false, Zr[c], (short)0, y,  false, false);
        #pragma unroll
        for (int c = 0; c < 4; ++c)
            zn = __builtin_amdgcn_wmma_f32_16x16x4_f32(false, Greg[c], false, Xr[c], (short)0, zn, false, false);
        #pragma unroll
        for (int c = 0; c < 4; ++c)
            zn = __builtin_amdgcn_wmma_f32_16x16x4_f32(false, Preg[c], false, Zr[c], (short)0, zn, false, false);

        // write back state: D-layout (M=r+8*half, N=m16) -> zt[seq][state]
        v4f zlo = __builtin_shufflevector(zn, zn, 0, 1, 2, 3);
        v4f zhi = __builtin_shufflevector(zn, zn, 4, 5, 6, 7);
        *(v4f*)&ztw[m16 * 16 + half * 8]     = zlo;
        *(v4f*)&ztw[m16 * 16 + half * 8 + 4] = zhi;

        // clamp + stage Y into [seq][t][f]
        #pragma unroll
        for (int r = 0; r < 8; ++r) {
            float yv = fminf(fmaxf(y[r], -1.f), 1.f);
            yt[m16 * 144 + (r + 8 * half) * 9 + w] = yv;
        }
        if (w == 0 && blk + 1 < NBLK) asm volatile("s_wait_asynccnt 0x0" ::: "memory");
        __syncthreads();  // ytile complete

        // cooperative coalesced store: per seq, 16t*9f = 144 contiguous floats
        int valid = (blk == NBLK - 1) ? 8 : 16;
        int t0 = blk * LBLK;
        #pragma unroll
        for (int k2 = 0; k2 < 8; ++k2) {
            int flat = tid + k2 * 288;           // 0..2303
            int s  = flat / 144;
            int r  = flat - s * 144;
            int tl = r / 9;
            if (tl < valid)
                out[((size_t)(seq_base + s) * TT + t0 + tl) * 9 + (r - tl * 9)] = yt[flat];
        }
    }
}

extern "C" void kernel_launch(void* const* d_in, const int* in_sizes, int n_in,
                              void* d_out, int out_size, void* d_ws, size_t ws_size,
                              hipStream_t stream) {
    const float* x  = (const float*)d_in[0];
    const float* Bc = (const float*)d_in[1];
    const float* Ac = (const float*)d_in[2];
    float* out = (float*)d_out;
    int K = in_sizes[1] / NF;   // padded tap count (a[0]==1)
    // 4096 sequences / 16 per workgroup; 9 waves (one per filter) per workgroup
    fb_wmma_kernel<<<256, 288, 0, stream>>>(x, Bc, Ac, out, K);
}